// RecurrentActorCritic_13632226197664
// MI455X (gfx1250) — compile-verified
//
#include <hip/hip_runtime.h>
#include <math.h>

// Problem constants (match the reference).
#define TT   8192
#define OBSD 512
#define AD   18
#define HID  1024
#define G4   (4 * HID)

typedef __attribute__((ext_vector_type(2))) float v2f;
typedef __attribute__((ext_vector_type(8))) float v8f;

// ---------------------------------------------------------------------------
// Register-blocked NT GEMM via V_WMMA_F32_16X16X4_F32:
//   C[M,N] = A[M,K] * B[N,K]^T + bias0 (+bias1) (+tanh)
// One wave owns an (MT*16) x (NT*16) macro-tile: MT*NT v8f accumulators.
// Per k-chunk (K step 4) it loads MT A-fragments + NT B-fragments and issues
// MT*NT independent WMMAs -> 4x the arithmetic intensity of 1x1 tiling and
// hazard-free back-to-back WMMA issue.
// A-operand layout (ISA 7.12.2, 32-bit A 16x4): lanes 0-15 hold M=lane,
// VGPR0/1 = K0/K1; lanes 16-31 = K2/K3.  C f32 layout: VGPR r -> M = r + 8*lh.
// ---------------------------------------------------------------------------
template <int ACT, int MT, int NT>  // ACT: 0 = none, 1 = tanh
__global__ __launch_bounds__(32) void wmma_gemm_nt(
    const float* __restrict__ A, const float* __restrict__ B,
    const float* __restrict__ bias0, const float* __restrict__ bias1,
    float* __restrict__ C, int M, int N, int K) {
  const int tileN = blockIdx.x, tileM = blockIdx.y;
  const int lane = threadIdx.x;
  const int lh   = lane >> 4;   // which half of the wave
  const int l16  = lane & 15;

  const float* Ap[MT];
  const float* Bp[NT];
#pragma unroll
  for (int mi = 0; mi < MT; ++mi)
    Ap[mi] = A + (size_t)(tileM * (16 * MT) + mi * 16 + l16) * K;
#pragma unroll
  for (int ni = 0; ni < NT; ++ni)
    Bp[ni] = B + (size_t)(tileN * (16 * NT) + ni * 16 + l16) * K;

  v8f acc[MT][NT];
#pragma unroll
  for (int mi = 0; mi < MT; ++mi)
#pragma unroll
    for (int ni = 0; ni < NT; ++ni) acc[mi][ni] = (v8f){};

  for (int k = 0; k < K; k += 4) {
    const int kk = k + lh * 2;
    v2f a[MT], b[NT];
#pragma unroll
    for (int mi = 0; mi < MT; ++mi) { a[mi].x = Ap[mi][kk]; a[mi].y = Ap[mi][kk + 1]; }
#pragma unroll
    for (int ni = 0; ni < NT; ++ni) { b[ni].x = Bp[ni][kk]; b[ni].y = Bp[ni][kk + 1]; }
#pragma unroll
    for (int mi = 0; mi < MT; ++mi)
#pragma unroll
      for (int ni = 0; ni < NT; ++ni)
        // 8 args: (neg_a, A, neg_b, B, c_mod, C, reuse_a, reuse_b)
        acc[mi][ni] = __builtin_amdgcn_wmma_f32_16x16x4_f32(
            false, a[mi], false, b[ni], (short)0, acc[mi][ni], false, false);
  }

#pragma unroll
  for (int ni = 0; ni < NT; ++ni) {
    const int n = tileN * (16 * NT) + ni * 16 + l16;
    float bv = bias0[n];
    if (bias1) bv += bias1[n];
#pragma unroll
    for (int mi = 0; mi < MT; ++mi) {
#pragma unroll
      for (int r = 0; r < 8; ++r) {
        const int m = tileM * (16 * MT) + mi * 16 + lh * 8 + r;
        float v = acc[mi][ni][r] + bv;
        if (ACT == 1) v = tanhf(v);
        C[(size_t)m * N + n] = v;
      }
    }
  }
}

// ---------------------------------------------------------------------------
// Pack [Wp(18,1024); Wv(1,1024); zeros] into a 32x1024 head weight + 32 bias,
// so the policy/value head becomes one WMMA-tileable GEMM.
// ---------------------------------------------------------------------------
__global__ void pack_head_kernel(const float* __restrict__ Wp,
                                 const float* __restrict__ bp,
                                 const float* __restrict__ Wv,
                                 const float* __restrict__ bv,
                                 float* __restrict__ Wpad,
                                 float* __restrict__ bpad) {
  const int i = blockIdx.x * blockDim.x + threadIdx.x;
  if (i < 32 * HID) {
    const int r = i >> 10, k = i & (HID - 1);
    float v = 0.f;
    if (r < AD)       v = Wp[r * HID + k];
    else if (r == AD) v = Wv[k];
    Wpad[i] = v;
  }
  if (i < 32) {
    float v = 0.f;
    if (i < AD)       v = bp[i];
    else if (i == AD) v = bv[0];
    bpad[i] = v;
  }
}

// ---------------------------------------------------------------------------
// Sequential LSTM scan. 128 persistent blocks x 256 threads = 1024 waves.
// Wave w owns h-index w; it computes the 4 gate rows (j, H+j, 2H+j, 3H+j)
// of W_hh · h with b128 loads + butterfly reduction, then the cell update.
// h is double-buffered in global memory; one atomic grid barrier per step.
// h2 (pre-reset) is stored to H2[t] for the post-hoc head GEMM.
// ---------------------------------------------------------------------------
__device__ __forceinline__ float sigmoidf_(float x) {
  return 1.f / (1.f + __expf(-x));
}

__global__ __launch_bounds__(256) void lstm_scan_kernel(
    const float* __restrict__ GX, const float* __restrict__ Whh,
    const float* __restrict__ dones, float* __restrict__ H2,
    float* __restrict__ hbuf /* [2*HID] */, float* __restrict__ cbuf /* [HID] */,
    unsigned* __restrict__ bar, int nT) {
  __shared__ __align__(16) float hs[HID];
  const int lane = threadIdx.x & 31;
  const int j = (blockIdx.x * blockDim.x + threadIdx.x) >> 5;  // 0..1023
  const unsigned nblk = gridDim.x;

  const float* wi = Whh + (size_t)j * HID;
  const float* wf = wi + (size_t)HID * HID;
  const float* wg = wf + (size_t)HID * HID;
  const float* wo = wg + (size_t)HID * HID;

  for (int t = 0; t < nT; ++t) {
    const float* hRead  = hbuf + (size_t)(t & 1) * HID;
    float*       hWrite = hbuf + (size_t)((t + 1) & 1) * HID;

    // Snapshot h into LDS once per block (4 KB, feeds 8 x 4 dot products).
    for (int k = threadIdx.x; k < HID; k += blockDim.x) hs[k] = hRead[k];
    __syncthreads();

    const float* gx = GX + (size_t)t * G4;
    const bool reset = dones[t] > 0.5f;

    __builtin_prefetch(wi, 0, 1);
    __builtin_prefetch(wf, 0, 1);
    __builtin_prefetch(wg, 0, 1);
    __builtin_prefetch(wo, 0, 1);
    float si = 0.f, sf = 0.f, sg = 0.f, so = 0.f;
    for (int k = lane * 4; k < HID; k += 128) {
      const float4 hv = *(const float4*)(hs + k);
      float4 w = *(const float4*)(wi + k);
      si += w.x * hv.x + w.y * hv.y + w.z * hv.z + w.w * hv.w;
      w = *(const float4*)(wf + k);
      sf += w.x * hv.x + w.y * hv.y + w.z * hv.z + w.w * hv.w;
      w = *(const float4*)(wg + k);
      sg += w.x * hv.x + w.y * hv.y + w.z * hv.z + w.w * hv.w;
      w = *(const float4*)(wo + k);
      so += w.x * hv.x + w.y * hv.y + w.z * hv.z + w.w * hv.w;
    }
#pragma unroll
    for (int off = 16; off; off >>= 1) {
      si += __shfl_xor(si, off);
      sf += __shfl_xor(sf, off);
      sg += __shfl_xor(sg, off);
      so += __shfl_xor(so, off);
    }
    const float iv = sigmoidf_(gx[j] + si);
    const float fv = sigmoidf_(gx[HID + j] + sf);
    const float gv = tanhf(gx[2 * HID + j] + sg);
    const float ov = sigmoidf_(gx[3 * HID + j] + so);
    const float c2 = fv * cbuf[j] + iv * gv;  // c owned exclusively by this wave
    const float h2 = ov * tanhf(c2);

    if (lane == 0) {
      H2[(size_t)t * HID + j] = h2;           // pre-reset h2 feeds the head
      hWrite[j] = reset ? 0.f : h2;
      cbuf[j]   = reset ? 0.f : c2;
    }
    // Release writes, then lock-step grid barrier (monotone counter, no ABA).
    __threadfence();
    __syncthreads();
    if (threadIdx.x == 0) {
      atomicAdd(bar, 1u);
      const unsigned target = nblk * (unsigned)(t + 1);
      while (atomicAdd(bar, 0u) < target) __builtin_amdgcn_s_sleep(2);
    }
    __syncthreads();
    __threadfence();  // acquire: don't read stale h lines from near caches
  }
}

// ---------------------------------------------------------------------------
// Per-timestep log-softmax over 18 logits + entropy + value pass-through.
// ---------------------------------------------------------------------------
__global__ void finalize_kernel(const float* __restrict__ LOGI /* [T,32] */,
                                const int* __restrict__ actions,
                                float* __restrict__ out, int nT) {
  const int t = blockIdx.x * blockDim.x + threadIdx.x;
  if (t >= nT) return;
  const float* row = LOGI + (size_t)t * 32;
  float mx = -INFINITY;
#pragma unroll
  for (int a = 0; a < AD; ++a) mx = fmaxf(mx, row[a]);
  float se = 0.f;
#pragma unroll
  for (int a = 0; a < AD; ++a) se += __expf(row[a] - mx);
  const float lse = mx + __logf(se);
  float ent = 0.f;
#pragma unroll
  for (int a = 0; a < AD; ++a) {
    const float ls = row[a] - lse;
    ent -= __expf(ls) * ls;
  }
  out[t]          = row[actions[t]] - lse;  // logprob
  out[nT + t]     = ent;                    // entropy
  out[2 * nT + t] = row[AD];                // value
}

// ---------------------------------------------------------------------------
extern "C" void kernel_launch(void* const* d_in, const int* in_sizes, int n_in,
                              void* d_out, int out_size, void* d_ws,
                              size_t ws_size, hipStream_t stream) {
  (void)in_sizes; (void)n_in; (void)out_size; (void)ws_size;
  const float* obs   = (const float*)d_in[0];
  const int*   acts  = (const int*)d_in[1];
  const float* dones = (const float*)d_in[2];
  const float* encW  = (const float*)d_in[3];
  const float* encB  = (const float*)d_in[4];
  const float* Wih   = (const float*)d_in[5];
  const float* Whh   = (const float*)d_in[6];
  const float* bih   = (const float*)d_in[7];
  const float* bhh   = (const float*)d_in[8];
  const float* Wp    = (const float*)d_in[9];
  const float* bp    = (const float*)d_in[10];
  const float* Wv    = (const float*)d_in[11];
  const float* bvv   = (const float*)d_in[12];
  float* out = (float*)d_out;

  // Workspace layout (fp32 unless noted), ~202 MB total.
  float* ws   = (float*)d_ws;
  float* X    = ws;                         // [T, H]
  float* GX   = X + (size_t)TT * HID;       // [T, 4H]
  float* H2   = GX + (size_t)TT * G4;       // [T, H]
  float* LOGI = H2 + (size_t)TT * HID;      // [T, 32]
  float* WPAD = LOGI + (size_t)TT * 32;     // [32, H]
  float* BPAD = WPAD + (size_t)32 * HID;    // [32]
  float* HBUF = BPAD + 32;                  // [2, H]  double-buffered h
  float* CBUF = HBUF + 2 * HID;             // [H]
  unsigned* BAR = (unsigned*)(CBUF + HID);  // barrier counter

  // Zero h/c/barrier (contiguous region). Graph-capture-safe async memset.
  hipMemsetAsync(HBUF, 0, (size_t)(3 * HID) * sizeof(float) + 16 * sizeof(unsigned),
                 stream);

  pack_head_kernel<<<dim3((32 * HID + 255) / 256), dim3(256), 0, stream>>>(
      Wp, bp, Wv, bvv, WPAD, BPAD);

  // Phase 1: X = tanh(obs * enc_W^T + enc_b)      [8192 x 1024, K=512], 64x64 tiles
  wmma_gemm_nt<1, 4, 4><<<dim3(HID / 64, TT / 64), dim3(32), 0, stream>>>(
      obs, encW, encB, nullptr, X, TT, HID, OBSD);
  // Phase 2: GX = X * W_ih^T + (b_ih + b_hh)      [8192 x 4096, K=1024], 64x64 tiles
  wmma_gemm_nt<0, 4, 4><<<dim3(G4 / 64, TT / 64), dim3(32), 0, stream>>>(
      X, Wih, bih, bhh, GX, TT, G4, HID);
  // Phase 3: sequential LSTM scan producing H2.
  lstm_scan_kernel<<<dim3(128), dim3(256), 0, stream>>>(
      GX, Whh, dones, H2, HBUF, CBUF, BAR, TT);
  // Phase 4: head GEMM  LOGI = H2 * [Wp;Wv;0]^T   [8192 x 32, K=1024], 64x32 tiles
  wmma_gemm_nt<0, 4, 2><<<dim3(1, TT / 64), dim3(32), 0, stream>>>(
      H2, WPAD, BPAD, nullptr, LOGI, TT, 32, HID);
  // Phase 5: log-softmax / entropy / value.
  finalize_kernel<<<dim3((TT + 255) / 256), dim3(256), 0, stream>>>(
      LOGI, acts, out, TT);
}